// RGCN_85993835200954
// MI455X (gfx1250) — compile-verified
//
#include <hip/hip_runtime.h>
#include <hip/hip_bf16.h>
#include <math.h>

// ---------------------------------------------------------------------------
// Problem constants (match the reference)
// ---------------------------------------------------------------------------
constexpr int B       = 8;
constexpr int NO      = 1024;
constexpr int NM      = 256;
constexpr int IN_OPT  = 64;
constexpr int IN_MAS  = 64;
constexpr int OUT     = 32;
constexpr int HEADS   = 4;
constexpr int TYPES   = 3;
constexpr int BASES   = 3;
constexpr int HID     = 128;
constexpr int ROWS    = B * NO;        // 8192
constexpr int TOUT    = TYPES * OUT;   // 96
constexpr int PO      = HEADS * OUT;   // 128  (output feature dim)
constexpr float NEG_SLOPE = 0.2f;

// workspace layout (in floats)
constexpr int WS_S    = 0;                        // [2]  s1, s0 (fc_edge collapse)
constexpr int WS_WCAT = 16;                       // [64][96]  basis-combined W
constexpr int WS_P    = WS_WCAT + IN_MAS * TOUT;  // [8192][128] fc_w0 output
constexpr int WS_G    = WS_P + ROWS * PO;         // [8192][64]  aggregated mach feats

typedef __attribute__((ext_vector_type(2))) float v2f;
typedef __attribute__((ext_vector_type(8))) float v8f;

// fp32 WMMA: D(16x16) = A(16x4) * B(4x16) + C ; exact fp32 FMA semantics (RNE)
#define WMMA_F32(a, b, c) \
  __builtin_amdgcn_wmma_f32_16x16x4_f32(false, (a), false, (b), (short)0, (c), false, false)

__device__ __forceinline__ float elu1(float x) {
  return x > 0.f ? x : (__expf(x) - 1.f);
}

// ---------------------------------------------------------------------------
// K0: tiny prep. s1 = we1·we2 ; s0 = be1·we2 + be2 ; Wcat = coeff@v reshaped.
// ---------------------------------------------------------------------------
__global__ void k0_prep(const float* __restrict__ we1, const float* __restrict__ be1,
                        const float* __restrict__ we2, const float* __restrict__ be2,
                        const float* __restrict__ v,   const float* __restrict__ coeff,
                        float* __restrict__ ws) {
  __shared__ float red[128];
  const int t = threadIdx.x;  // 128 threads
  const float p1 = we1[t] * we2[t];
  const float p0 = be1[t] * we2[t];

  red[t] = p1;
  __syncthreads();
  for (int s = 64; s > 0; s >>= 1) { if (t < s) red[t] += red[t + s]; __syncthreads(); }
  if (t == 0) ws[WS_S + 0] = red[0];
  __syncthreads();

  red[t] = p0;
  __syncthreads();
  for (int s = 64; s > 0; s >>= 1) { if (t < s) red[t] += red[t + s]; __syncthreads(); }
  if (t == 0) ws[WS_S + 1] = red[0] + be2[0];

  // Wcat[i][ty*OUT + j] = sum_r coeff[ty][r] * v[r][i][j]
  for (int idx = t; idx < IN_MAS * TOUT; idx += 128) {
    const int i  = idx / TOUT;
    const int c  = idx % TOUT;
    const int ty = c / OUT;
    const int j  = c % OUT;
    float acc = 0.f;
    #pragma unroll
    for (int r = 0; r < BASES; ++r)
      acc += coeff[ty * BASES + r] * v[(r * IN_MAS + i) * OUT + j];
    ws[WS_WCAT + idx] = acc;
  }
}

// ---------------------------------------------------------------------------
// K1: fc_w0.  P = elu(X @ W1 + b1) @ W2 + b2.   One wave per 16-row tile.
// ---------------------------------------------------------------------------
__global__ void __launch_bounds__(32)
k1_fcw0(const float* __restrict__ X,  const float* __restrict__ W1,
        const float* __restrict__ b1, const float* __restrict__ W2,
        const float* __restrict__ b2, float* __restrict__ ws) {
  __shared__ float Hs[16][HID + 4];

  const int lane = threadIdx.x;
  const int m    = lane & 15;
  const int half = lane >> 4;
  const int r0   = blockIdx.x * 16;

  const v8f zero8 = {0.f, 0.f, 0.f, 0.f, 0.f, 0.f, 0.f, 0.f};

  // ---- stage 1: H(16x128) = elu(X(16x64) @ W1(64x128) + b1) ----
  v8f acc[8];
  #pragma unroll
  for (int t = 0; t < 8; ++t) acc[t] = zero8;

  for (int k = 0; k < IN_OPT; k += 4) {
    const int kk = k + 2 * half;
    const v2f a = *(const v2f*)(X + (size_t)(r0 + m) * IN_OPT + kk);  // 8B aligned
    #pragma unroll
    for (int t = 0; t < 8; ++t) {
      const int n = t * 16 + m;
      v2f bb;
      bb.x = W1[(size_t)kk * HID + n];
      bb.y = W1[(size_t)(kk + 1) * HID + n];
      acc[t] = WMMA_F32(a, bb, acc[t]);
    }
  }

  #pragma unroll
  for (int t = 0; t < 8; ++t) {
    const int n = t * 16 + m;
    const float bias = b1[n];
    #pragma unroll
    for (int r = 0; r < 8; ++r)
      Hs[r + 8 * half][n] = elu1(acc[t][r] + bias);   // C-layout -> LDS row-major
  }
  __syncthreads();

  // ---- stage 2: P(16x128) = H @ W2(128x128) + b2 ----
  v8f acc2[8];
  #pragma unroll
  for (int t = 0; t < 8; ++t) acc2[t] = zero8;

  for (int k = 0; k < HID; k += 4) {
    const int kk = k + 2 * half;
    v2f a;
    a.x = Hs[m][kk];
    a.y = Hs[m][kk + 1];
    #pragma unroll
    for (int t = 0; t < 8; ++t) {
      const int n = t * 16 + m;
      v2f bb;
      bb.x = W2[(size_t)kk * HID + n];
      bb.y = W2[(size_t)(kk + 1) * HID + n];
      acc2[t] = WMMA_F32(a, bb, acc2[t]);
    }
  }

  float* __restrict__ P = ws + WS_P;
  #pragma unroll
  for (int t = 0; t < 8; ++t) {
    const int n = t * 16 + m;
    const float bias = b2[n];
    #pragma unroll
    for (int r = 0; r < 8; ++r)
      P[(size_t)(r0 + r + 8 * half) * PO + n] = acc2[t][r] + bias;
  }
}

// ---------------------------------------------------------------------------
// K2: g[b] = (h_edg*adj)(1024x256) @ feat_mas[b](256x64).
//     A synthesized on the fly: e = adj? fe*s1+s0 : 0.  One wave / 16-row tile.
// ---------------------------------------------------------------------------
__global__ void __launch_bounds__(32)
k2_gather(const int* __restrict__ adj, const float* __restrict__ fe,
          const float* __restrict__ fmas, float* __restrict__ ws) {
  const int lane = threadIdx.x;
  const int m    = lane & 15;
  const int half = lane >> 4;
  const int b    = blockIdx.y;
  const int o0   = blockIdx.x * 16;

  const float s1 = ws[WS_S + 0];
  const float s0 = ws[WS_S + 1];

  const v8f zero8 = {0.f, 0.f, 0.f, 0.f, 0.f, 0.f, 0.f, 0.f};
  v8f acc[4];
  #pragma unroll
  for (int t = 0; t < 4; ++t) acc[t] = zero8;

  const size_t rowg = (size_t)b * NO + o0 + m;
  const int*   ap   = adj + rowg * NM;
  const float* ep   = fe  + rowg * NM;

  for (int k = 0; k < NM; k += 4) {
    const int kk = k + 2 * half;
    v2f a;
    a.x = (ap[kk]     == 1) ? (ep[kk]     * s1 + s0) : 0.f;
    a.y = (ap[kk + 1] == 1) ? (ep[kk + 1] * s1 + s0) : 0.f;
    #pragma unroll
    for (int t = 0; t < 4; ++t) {
      const int n = t * 16 + m;
      v2f bb;
      bb.x = fmas[((size_t)b * NM + kk)     * IN_MAS + n];
      bb.y = fmas[((size_t)b * NM + kk + 1) * IN_MAS + n];
      acc[t] = WMMA_F32(a, bb, acc[t]);
    }
  }

  float* __restrict__ g = ws + WS_G;
  #pragma unroll
  for (int t = 0; t < 4; ++t) {
    const int n = t * 16 + m;
    #pragma unroll
    for (int r = 0; r < 8; ++r)
      g[((size_t)b * NO + o0 + r + 8 * half) * IN_MAS + n] = acc[t][r];
  }
}

// ---------------------------------------------------------------------------
// K3: fm = g @ Wcat ; s = elu(fm@wom1+bom1)@wom2+bom2 ; out = lrelu(P + s).
// ---------------------------------------------------------------------------
__global__ void __launch_bounds__(32)
k3_gate(const float* __restrict__ Wom1, const float* __restrict__ bom1,
        const float* __restrict__ wom2, const float* __restrict__ bom2,
        float* __restrict__ ws, float* __restrict__ outp) {
  __shared__ float Fs[16][TOUT + 4];   // fm tile, row-major
  __shared__ float red[32][8];
  __shared__ float srow[16];

  const int lane = threadIdx.x;
  const int m    = lane & 15;
  const int half = lane >> 4;
  const int r0   = blockIdx.x * 16;

  const float* __restrict__ g    = ws + WS_G;
  const float* __restrict__ Wcat = ws + WS_WCAT;
  const v8f zero8 = {0.f, 0.f, 0.f, 0.f, 0.f, 0.f, 0.f, 0.f};

  // ---- fm(16x96) = g(16x64) @ Wcat(64x96) ----
  v8f fa[6];
  #pragma unroll
  for (int t = 0; t < 6; ++t) fa[t] = zero8;

  for (int k = 0; k < IN_MAS; k += 4) {
    const int kk = k + 2 * half;
    const v2f a = *(const v2f*)(g + (size_t)(r0 + m) * IN_MAS + kk);
    #pragma unroll
    for (int t = 0; t < 6; ++t) {
      const int n = t * 16 + m;
      v2f bb;
      bb.x = Wcat[(size_t)kk * TOUT + n];
      bb.y = Wcat[(size_t)(kk + 1) * TOUT + n];
      fa[t] = WMMA_F32(a, bb, fa[t]);
    }
  }
  #pragma unroll
  for (int t = 0; t < 6; ++t) {
    const int n = t * 16 + m;
    #pragma unroll
    for (int r = 0; r < 8; ++r)
      Fs[r + 8 * half][n] = fa[t][r];
  }
  __syncthreads();

  // ---- u = elu(fm @ wom1 + bom1); ps[r] = sum_n u*wom2 (partial, per lane) ----
  float ps[8];
  #pragma unroll
  for (int r = 0; r < 8; ++r) ps[r] = 0.f;

  for (int t = 0; t < 8; ++t) {
    v8f u = zero8;
    for (int k = 0; k < TOUT; k += 4) {
      const int kk = k + 2 * half;
      v2f a;
      a.x = Fs[m][kk];
      a.y = Fs[m][kk + 1];
      const int n = t * 16 + m;
      v2f bb;
      bb.x = Wom1[(size_t)kk * HID + n];
      bb.y = Wom1[(size_t)(kk + 1) * HID + n];
      u = WMMA_F32(a, bb, u);
    }
    const int n = t * 16 + m;
    const float bias = bom1[n];
    const float w2   = wom2[n];
    #pragma unroll
    for (int r = 0; r < 8; ++r)
      ps[r] += elu1(u[r] + bias) * w2;
  }

  #pragma unroll
  for (int r = 0; r < 8; ++r) red[lane][r] = ps[r];
  __syncthreads();

  if (lane < 16) {
    // rows 0..7 live in lanes 0..15 (r=row), rows 8..15 in lanes 16..31 (r=row-8)
    const int rr = m & 7;
    const int hh = m >> 3;
    float s = 0.f;
    #pragma unroll
    for (int l = 0; l < 16; ++l) s += red[hh * 16 + l][rr];
    srow[m] = s + bom2[0];
  }
  __syncthreads();

  // ---- out = leaky_relu(P + s) ----
  const float* __restrict__ P = ws + WS_P;
  for (int idx = lane; idx < 16 * PO; idx += 32) {
    const int rr = idx / PO;
    const int n  = idx % PO;
    const float val = P[(size_t)(r0 + rr) * PO + n] + srow[rr];
    outp[(size_t)(r0 + rr) * PO + n] = val > 0.f ? val : NEG_SLOPE * val;
  }
}

// ---------------------------------------------------------------------------
extern "C" void kernel_launch(void* const* d_in, const int* in_sizes, int n_in,
                              void* d_out, int out_size, void* d_ws, size_t ws_size,
                              hipStream_t stream) {
  (void)in_sizes; (void)n_in; (void)out_size; (void)ws_size;

  const int*   adj       = (const int*)  d_in[0];
  /* d_in[1] = batch_idxes: unused by the reference */
  const float* feat_opt  = (const float*)d_in[2];
  const float* feat_mas  = (const float*)d_in[3];
  const float* feat_edge = (const float*)d_in[4];
  const float* w0_w1     = (const float*)d_in[5];
  const float* w0_b1     = (const float*)d_in[6];
  const float* w0_w2     = (const float*)d_in[7];
  const float* w0_b2     = (const float*)d_in[8];
  const float* v         = (const float*)d_in[9];
  const float* coeff     = (const float*)d_in[10];
  const float* we1       = (const float*)d_in[11];
  const float* be1       = (const float*)d_in[12];
  const float* we2       = (const float*)d_in[13];
  const float* be2       = (const float*)d_in[14];
  const float* wom1      = (const float*)d_in[15];
  const float* bom1      = (const float*)d_in[16];
  const float* wom2      = (const float*)d_in[17];
  const float* bom2      = (const float*)d_in[18];

  float* ws  = (float*)d_ws;
  float* out = (float*)d_out;

  k0_prep<<<1, 128, 0, stream>>>(we1, be1, we2, be2, v, coeff, ws);
  k1_fcw0<<<ROWS / 16, 32, 0, stream>>>(feat_opt, w0_w1, w0_b1, w0_w2, w0_b2, ws);
  dim3 g2(NO / 16, B);
  k2_gather<<<g2, 32, 0, stream>>>(adj, feat_edge, feat_mas, ws);
  k3_gate<<<ROWS / 16, 32, 0, stream>>>(wom1, bom1, wom2, bom2, ws, out);
}